// VQEmbeddingEMA_63402307224290
// MI455X (gfx1250) — compile-verified
//
#include <hip/hip_runtime.h>
#include <hip/hip_bf16.h>

typedef float v2f __attribute__((ext_vector_type(2)));
typedef float v4f __attribute__((ext_vector_type(4)));
typedef float v8f __attribute__((ext_vector_type(8)));

#define N_CB  8
#define M_EMB 1024
#define D_EMB 128
#define H_LAT 64
#define B_SZ  256
#define L_PER (B_SZ * H_LAT)   // 16384 rows per codebook
#define TL    128              // L rows per block
#define MC    64               // embedding rows staged per chunk
#define NCHUNK (M_EMB / MC)    // 16
#define LSTR  132              // padded LDS row stride (floats) -> bank-conflict-free b64 loads

#define ZQ_ELEMS (B_SZ * N_CB * D_EMB * H_LAT)   // 16777216
#define OUT_LOSS ((size_t)ZQ_ELEMS)
#define OUT_ENT  ((size_t)ZQ_ELEMS + 1)
#define OUT_PERP ((size_t)ZQ_ELEMS + 2)
#define OUT_IDX  ((size_t)ZQ_ELEMS + 3)

// ---- CDNA5 async global->LDS path (ASYNCcnt-tracked DMA), with sync fallback ----
#if defined(__gfx1250__) && __has_builtin(__builtin_amdgcn_global_load_async_to_lds_b128)
#define HAS_ASYNC 1
#else
#define HAS_ASYNC 0
#endif

#if HAS_ASYNC
// exact parameter types per hipcc diagnostic:
//   arg0: int __vector(4) addrspace(1)*   (global source)
//   arg1: int __vector(4) addrspace(3)*   (LDS destination)
typedef int v4i_g __attribute__((vector_size(16)));
typedef __attribute__((address_space(1))) v4i_g* as1_v4p;
typedef __attribute__((address_space(3))) v4i_g* as3_v4p;
#endif

__device__ __forceinline__ void cp16_to_lds(const float* g, float* l) {
#if HAS_ASYNC
    __builtin_amdgcn_global_load_async_to_lds_b128((as1_v4p)g, (as3_v4p)l, 0, 0);
#else
    *(v4f*)l = *(const v4f*)g;
#endif
}

__device__ __forceinline__ void wait_async0() {
#if HAS_ASYNC
#if __has_builtin(__builtin_amdgcn_s_wait_asynccnt)
    __builtin_amdgcn_s_wait_asynccnt(0);
#else
    asm volatile("s_wait_asynccnt 0x0" ::: "memory");
#endif
#endif
}

// workspace layout (bytes):
//   [0, 32768)      hist   : 8*1024 u32
//   [32768, 32772)  loss   : f32 accumulator
//   [33024, 65792)  esq    : 8*1024 f32 (||e||^2 per code)

__global__ void vq_prep(const float* __restrict__ emb, unsigned* __restrict__ hist,
                        float* __restrict__ loss_acc, float* __restrict__ esq) {
    int t = blockIdx.x * blockDim.x + threadIdx.x;
    if (t < N_CB * M_EMB) {
        hist[t] = 0u;
        const v4f* row = (const v4f*)(emb + (size_t)t * D_EMB);
        float s = 0.f;
        #pragma unroll
        for (int i = 0; i < D_EMB / 4; ++i) {
            v4f v = row[i];
            s += v.x * v.x + v.y * v.y + v.z * v.z + v.w * v.w;
        }
        esq[t] = s;
    }
    if (t == N_CB * M_EMB) *loss_acc = 0.f;
}

__global__ __launch_bounds__(256)
void vq_main(const float* __restrict__ x, const float* __restrict__ emb,
             const float* __restrict__ esq, unsigned* __restrict__ hist,
             float* __restrict__ loss_acc, float* __restrict__ out) {
    __shared__ float xs[TL * LSTR];          // 67584 B : x tile [l_local][d]
    __shared__ float es[2][MC * LSTR];       // 2 x 33792 B : double-buffered embedding chunk
    __shared__ float esq_s[M_EMB];           // 4096 B : ||e||^2 for whole codebook
    __shared__ int   idx_s[TL];
    __shared__ float red[8];

    const int n    = blockIdx.y;
    const int l0   = blockIdx.x * TL;        // multiple of 128
    const int b0   = l0 >> 6;                // covers b0, b0+1 (h spans full 0..63)
    const int tid  = threadIdx.x;
    const int lane = tid & 31;
    const int w    = tid >> 5;
    const int half = lane >> 4;
    const int l15  = lane & 15;

    const float* embn = emb + (size_t)n * M_EMB * D_EMB;

    // issue async DMA of chunk 0 first so it overlaps the x-tile staging
    {
        #pragma unroll
        for (int it = 0; it < 8; ++it) {
            int t  = it * 256 + tid;
            int d4 = t & 31;
            int j  = t >> 5;
            cp16_to_lds(embn + ((size_t)j * D_EMB + d4 * 4), &es[0][j * LSTR + d4 * 4]);
        }
    }

    // ---- stage x tile: x[b, n, d, h]; rows contiguous over h, transpose into [l][d] ----
    #pragma unroll
    for (int it = 0; it < 16; ++it) {
        int t  = it * 256 + tid;
        int h4 = t & 15;
        int d  = (t >> 4) & 127;
        int bl = t >> 11;
        v4f v = *(const v4f*)(x + ((((size_t)(b0 + bl) * N_CB + n) * D_EMB + d) * H_LAT + h4 * 4));
        int rbase = bl * 64 + h4 * 4;
        xs[(rbase + 0) * LSTR + d] = v.x;
        xs[(rbase + 1) * LSTR + d] = v.y;
        xs[(rbase + 2) * LSTR + d] = v.z;
        xs[(rbase + 3) * LSTR + d] = v.w;
    }
    // ---- stage ||e||^2 for the whole codebook once ----
    #pragma unroll
    for (int i = 0; i < M_EMB / 256; ++i)
        esq_s[i * 256 + tid] = esq[n * M_EMB + i * 256 + tid];

    wait_async0();
    __syncthreads();

    // ---- hoist full A operand (16 rows x 128 K) into registers, WMMA layout ----
    const int row_a = w * 16 + l15;
    const float* xrow = &xs[row_a * LSTR + half * 2];
    v2f aA[32];
    #pragma unroll
    for (int k = 0; k < 32; ++k) aA[k] = *(const v2f*)(xrow + k * 4);

    float minv[8];
    int   mini[8];
    #pragma unroll
    for (int v = 0; v < 8; ++v) { minv[v] = 3.0e38f; mini[v] = 0; }

    for (int c = 0; c < NCHUNK; ++c) {
        const int buf = c & 1;
        // prefetch chunk c+1 into the other buffer (consumers of it finished at
        // the barrier that ended iteration c-1, so the buffer is free)
        if (c + 1 < NCHUNK) {
            #pragma unroll
            for (int it = 0; it < 8; ++it) {
                int t  = it * 256 + tid;
                int d4 = t & 31;
                int j  = t >> 5;
                cp16_to_lds(embn + (((size_t)(c + 1) * MC + j) * D_EMB + d4 * 4),
                            &es[buf ^ 1][j * LSTR + d4 * 4]);
            }
        }

        v8f zero = {};
        v8f acc[4];
        #pragma unroll
        for (int mt = 0; mt < 4; ++mt) acc[mt] = zero;

        const float* brow[4];
        #pragma unroll
        for (int mt = 0; mt < 4; ++mt) brow[mt] = &es[buf][(mt * 16 + l15) * LSTR + half * 2];

        #pragma unroll
        for (int k = 0; k < 32; ++k) {
            v2f a = aA[k];
            #pragma unroll
            for (int mt = 0; mt < 4; ++mt) {
                v2f b = *(const v2f*)(brow[mt] + k * 4);
                acc[mt] = __builtin_amdgcn_wmma_f32_16x16x4_f32(
                    false, a, false, b, (short)0, acc[mt], false, false);
            }
        }

        // epilogue: dist = ||e||^2 - 2*cross  (x_sq row-constant, irrelevant to argmin)
        #pragma unroll
        for (int mt = 0; mt < 4; ++mt) {
            int   m_col = c * MC + mt * 16 + l15;
            float eq    = esq_s[c * MC + mt * 16 + l15];
            #pragma unroll
            for (int v = 0; v < 8; ++v) {
                float dv = eq - 2.0f * acc[mt][v];
                if (dv < minv[v]) { minv[v] = dv; mini[v] = m_col; }
            }
        }

        if (c + 1 < NCHUNK) {
            wait_async0();       // my wave's DMA for chunk c+1 landed
            __syncthreads();     // everyone's landed; chunk c fully consumed
        }
    }

    // ---- lexicographic argmin across the 16 lanes that share each C row ----
    #pragma unroll
    for (int v = 0; v < 8; ++v) {
        float bv = minv[v];
        int   bi = mini[v];
        #pragma unroll
        for (int s = 1; s < 16; s <<= 1) {
            float ov = __shfl_xor(bv, s, 32);
            int   oi = __shfl_xor(bi, s, 32);
            if (ov < bv || (ov == bv && oi < bi)) { bv = ov; bi = oi; }
        }
        if (l15 == 0) idx_s[w * 16 + half * 8 + v] = bi;
    }
    __syncthreads();

    // ---- indices output + histogram ----
    if (tid < TL) {
        int l = l0 + tid;
        int b = l >> 6, h = l & 63;
        int qi = idx_s[tid];
        atomicAdd(&hist[n * M_EMB + qi], 1u);
        out[OUT_IDX + ((size_t)b * N_CB + n) * H_LAT + h] = (float)qi;
    }

    // ---- z_q scatter + commitment-loss accumulation ----
    float lsum = 0.f;
    #pragma unroll 4
    for (int it = 0; it < 64; ++it) {
        int e  = it * 256 + tid;
        int h  = e & 63;
        int d  = (e >> 6) & 127;
        int bl = e >> 13;
        int li = bl * 64 + h;
        int qi = idx_s[li];
        float q  = embn[(size_t)qi * D_EMB + d];
        float xv = xs[li * LSTR + d];
        float df = xv - q;
        lsum += df * df;
        out[(((size_t)(b0 + bl) * N_CB + n) * D_EMB + d) * H_LAT + h] = q;
    }
    #pragma unroll
    for (int s = 1; s < 32; s <<= 1) lsum += __shfl_xor(lsum, s, 32);
    if (lane == 0) red[w] = lsum;
    __syncthreads();
    if (tid == 0) {
        float bs = 0.f;
        #pragma unroll
        for (int i = 0; i < 8; ++i) bs += red[i];
        atomicAdd(loss_acc, bs);
    }
}

__global__ void vq_final(const unsigned* __restrict__ hist,
                         const float* __restrict__ loss_acc,
                         float* __restrict__ out) {
    __shared__ float ent[8];
    int tid = threadIdx.x, lane = tid & 31, w = tid >> 5;   // wave w -> codebook w
    float s = 0.f;
    for (int m = lane; m < M_EMB; m += 32) {
        float p = (float)hist[w * M_EMB + m] * (1.0f / (float)L_PER);
        s += p * logf(p + 1e-10f);
    }
    #pragma unroll
    for (int d = 1; d < 32; d <<= 1) s += __shfl_xor(s, d, 32);
    if (lane == 0) ent[w] = -s;
    __syncthreads();
    if (tid == 0) {
        float em = 0.f, pm = 0.f;
        #pragma unroll
        for (int i = 0; i < 8; ++i) { em += ent[i]; pm += expf(ent[i]); }
        out[OUT_LOSS] = 0.25f * (*loss_acc) * (1.0f / (float)ZQ_ELEMS);
        out[OUT_ENT]  = em * 0.125f;
        out[OUT_PERP] = pm * 0.125f;
    }
}

extern "C" void kernel_launch(void* const* d_in, const int* in_sizes, int n_in,
                              void* d_out, int out_size, void* d_ws, size_t ws_size,
                              hipStream_t stream) {
    const float* x   = (const float*)d_in[0];
    const float* emb = (const float*)d_in[1];
    float* out = (float*)d_out;
    char* ws = (char*)d_ws;
    unsigned* hist  = (unsigned*)ws;
    float* loss_acc = (float*)(ws + 32768);
    float* esq      = (float*)(ws + 33024);

    vq_prep<<<(N_CB * M_EMB + 256) / 256 + 1, 256, 0, stream>>>(emb, hist, loss_acc, esq);
    dim3 grid(L_PER / TL, N_CB);
    vq_main<<<grid, 256, 0, stream>>>(x, emb, esq, hist, loss_acc, out);
    vq_final<<<1, 256, 0, stream>>>(hist, loss_acc, out);
}